// PytorchCanny_35493609734733
// MI455X (gfx1250) — compile-verified
//
#include <hip/hip_runtime.h>
#include <hip/hip_bf16.h>

#define WIMG 1920
#define HIMG 1080
#define TW 128
#define TH 32
// pass 1 (gauss+sobel) staging: halo 3 for input, halo 1 for smoothed
#define IN_H (TH + 6)
#define IN_W (TW + 6)
#define SM_H (TH + 2)
#define SM_W (TW + 2)
// pass 3 (hysteresis) staging: halo 2
#define HB (TH + 4)
#define WB (TW + 4)

typedef __attribute__((ext_vector_type(4))) unsigned int u32x4;
typedef __attribute__((ext_vector_type(4))) int          i32x4;
typedef __attribute__((ext_vector_type(8))) int          i32x8;

static __device__ __forceinline__ int imin(int a, int b) { return a < b ? a : b; }
static __device__ __forceinline__ int imax(int a, int b) { return a > b ? a : b; }

// ---------------------------------------------------------------------------
// Tensor Data Mover: 2D tile (tileH rows x tileW elems) from global -> LDS.
// Descriptor packed per CDNA5 D# layout (groups 0/1; groups 2/3 zero => 2D).
// tensor_dim == tile_dim so no OOB path is exercised (caller pre-clamps).
// LDS receives the tile densely packed (pitch == tileW elements).
// ---------------------------------------------------------------------------
static __device__ __forceinline__ void tdm_load_2d_to_lds(
    void* ldsDst, const void* gSrc,
    unsigned tileW, unsigned tileH, unsigned strideElems, unsigned dcode /*0=1B,2=4B*/)
{
  const unsigned ldsOff = (unsigned)(size_t)ldsDst;            // low 32 bits = LDS offset
  const unsigned long long ga = (unsigned long long)gSrc;

  u32x4 g0 = {
    1u,                                                        // count=1 (valid user D#)
    ldsOff,                                                    // lds_addr [63:32]
    (unsigned)(ga & 0xFFFFFFFFull),                            // global_addr lo
    (unsigned)((ga >> 32) & 0x01FFFFFFull) | (2u << 30)        // global_addr hi | type=2
  };
  i32x8 g1 = {
    (int)(dcode << 16),                                        // wg_mask=0, data_size
    (int)((tileW & 0xFFFFu) << 16),                            // tensor_dim0[15:0]
    (int)(((tileW >> 16) & 0xFFFFu) | ((tileH & 0xFFFFu) << 16)),
    (int)(((tileH >> 16) & 0xFFFFu) | ((tileW & 0xFFFFu) << 16)), // | tile_dim0
    (int)(tileH & 0xFFFFu),                                    // tile_dim1 (tile_dim2=0)
    (int)strideElems,                                          // tensor_dim0_stride lo32
    0, 0                                                       // stride hi / dim1_stride
  };
  i32x4 g2 = {0, 0, 0, 0};
  i32x4 g3 = {0, 0, 0, 0};
#if defined(__clang_major__) && (__clang_major__ >= 23)
  i32x8 g4 = {0, 0, 0, 0, 0, 0, 0, 0};
  __builtin_amdgcn_tensor_load_to_lds(g0, g1, g2, g3, g4, 0);
#else
  __builtin_amdgcn_tensor_load_to_lds(g0, g1, g2, g3, 0);
#endif
}

// ---------------------------------------------------------------------------
// Pass 1: x -> gaussian smooth -> sobel -> mag*255, angle deg % 180
// ---------------------------------------------------------------------------
__global__ __launch_bounds__(256) void canny_gauss_sobel(
    const float* __restrict__ x, const float* __restrict__ gw,
    const float* __restrict__ sxw, const float* __restrict__ syw,
    float* __restrict__ magOut, float* __restrict__ angOut)
{
  __shared__ float inT[IN_H * IN_W];
  __shared__ float smT[SM_H * SM_W];

  const int n  = blockIdx.z;
  const int x0 = blockIdx.x * TW, y0 = blockIdx.y * TH;
  const int sy = imax(y0 - 3, 0), ey = imin(y0 + TH + 3, HIMG);
  const int sx = imax(x0 - 3, 0), ex = imin(x0 + TW + 3, WIMG);
  const int rows = ey - sy, cols = ex - sx;
  const bool interior = (rows == IN_H) && (cols == IN_W);

  if (threadIdx.x < 32) {   // wave 0 issues the TDM transfer and waits on it
    const float* src = x + ((size_t)n * HIMG + sy) * (size_t)WIMG + sx;
    tdm_load_2d_to_lds(inT, src, (unsigned)cols, (unsigned)rows, WIMG, 2u);
    __builtin_amdgcn_s_wait_tensorcnt(0);
  }

  float g[25];
#pragma unroll
  for (int k = 0; k < 25; ++k) g[k] = gw[k] * (1.0f / 255.0f);  // fold the /255 in
  float kx[9], ky[9];
#pragma unroll
  for (int k = 0; k < 9; ++k) { kx[k] = sxw[k]; ky[k] = syw[k]; }

  __syncthreads();

  // smoothed over tile + 1 halo (zero outside image -> matches conv zero-pad)
  for (int i = threadIdx.x; i < SM_H * SM_W; i += 256) {
    const int r = i / SM_W, c = i % SM_W;
    const int gy = y0 - 1 + r, gx = x0 - 1 + c;
    float s = 0.0f;
    if (gy >= 0 && gy < HIMG && gx >= 0 && gx < WIMG) {
      if (interior) {
        const float* p = &inT[(r + 2) * IN_W + (c + 2)];
#pragma unroll
        for (int dy = -2; dy <= 2; ++dy)
#pragma unroll
          for (int dx = -2; dx <= 2; ++dx)
            s = fmaf(g[(dy + 2) * 5 + (dx + 2)], p[dy * IN_W + dx], s);
      } else {
        for (int dy = -2; dy <= 2; ++dy)
          for (int dx = -2; dx <= 2; ++dx) {
            const int iy = gy + dy, ix = gx + dx;
            if (iy >= 0 && iy < HIMG && ix >= 0 && ix < WIMG)
              s = fmaf(g[(dy + 2) * 5 + (dx + 2)], inT[(iy - sy) * cols + (ix - sx)], s);
          }
      }
    }
    smT[i] = s;
  }
  __syncthreads();

  for (int i = threadIdx.x; i < TH * TW; i += 256) {
    const int ty = i / TW, tx = i % TW;
    const int gy = y0 + ty, gx = x0 + tx;
    if (gy >= HIMG || gx >= WIMG) continue;
    const float* p = &smT[(ty + 1) * SM_W + (tx + 1)];
    float gxv = 0.0f, gyv = 0.0f;
#pragma unroll
    for (int dy = -1; dy <= 1; ++dy)
#pragma unroll
      for (int dx = -1; dx <= 1; ++dx) {
        const float v = p[dy * SM_W + dx];
        gxv = fmaf(kx[(dy + 1) * 3 + (dx + 1)], v, gxv);
        gyv = fmaf(ky[(dy + 1) * 3 + (dx + 1)], v, gyv);
      }
    const float mag = sqrtf(fmaf(gxv, gxv, fmaf(gyv, gyv, 1e-6f)));
    const float ang = atan2f(gyv, gxv);
    const size_t o = ((size_t)n * HIMG + gy) * (size_t)WIMG + gx;
    magOut[o] = mag * 255.0f;
    float ad = ang + 3.14159265358979323846f;
    ad = fmodf(ad, 6.28318530717958647692f);
    ad = ad * 57.295779513082320877f;     // degrees
    ad = fmodf(ad, 180.0f);
    angOut[o] = ad;
  }
}

// ---------------------------------------------------------------------------
// Pass 2: NMS with jnp.roll wrap semantics + double threshold -> packed byte
// bit0 = strong (>150 scaled), bit1 = weak ([50,150) scaled)
// ---------------------------------------------------------------------------
__global__ __launch_bounds__(256) void canny_nms(
    const float* __restrict__ magS, const float* __restrict__ angD,
    unsigned char* __restrict__ lh, int ntot)
{
  const long long gid = (long long)blockIdx.x * 256 + threadIdx.x;
  if (gid >= ntot) return;
  const int P = HIMG * WIMG;
  const int n = (int)(gid / P);
  const int rem = (int)(gid % P);
  const int y = rem / WIMG, xq = rem % WIMG;
  const float* m = magS + (size_t)n * P;
  const int yu = (y + HIMG - 1) % HIMG, yd = (y + 1) % HIMG;
  const int xl = (xq + WIMG - 1) % WIMG, xr = (xq + 1) % WIMG;
  const float c = m[(size_t)y * WIMG + xq];
  const float a = angD[gid];
  // 45-degree class; invariant under the %180 already applied to the angle
  const int cls = ((int)floorf(a * (1.0f / 45.0f) + 0.5f)) & 3;
  float t1, t2;
  if (cls == 0)      { t1 = m[(size_t)y  * WIMG + xr]; t2 = m[(size_t)y  * WIMG + xl]; }
  else if (cls == 1) { t1 = m[(size_t)yd * WIMG + xr]; t2 = m[(size_t)yu * WIMG + xq]; } // n_d1 & n_up (ref bug)
  else if (cls == 2) { t1 = m[(size_t)yd * WIMG + xq]; t2 = m[(size_t)yu * WIMG + xq]; }
  else               { t1 = m[(size_t)yd * WIMG + xl]; t2 = m[(size_t)yu * WIMG + xr]; } // n_d3 & n_d4
  const float sup = (c > t1 && c > t2) ? c : 0.0f;
  unsigned char b = 0;
  if (sup > 150.0f)                         b = 1;
  else if (sup >= 50.0f && sup < 150.0f)    b = 2;
  lh[gid] = b;
}

// ---------------------------------------------------------------------------
// Pass 3: both hysteresis iterations fused in LDS (halo 2, zero-pad) -> edges
// ---------------------------------------------------------------------------
__global__ __launch_bounds__(256) void canny_hyst(
    const unsigned char* __restrict__ lh, float* __restrict__ edges)
{
  __shared__ unsigned char inB[HB * WB];
  __shared__ unsigned char E0[HB * WB], L0[HB * WB], E1[HB * WB], L1[HB * WB];

  const int n  = blockIdx.z;
  const int x0 = blockIdx.x * TW, y0 = blockIdx.y * TH;
  const int sy = imax(y0 - 2, 0), ey = imin(y0 + TH + 2, HIMG);
  const int sx = imax(x0 - 2, 0), ex = imin(x0 + TW + 2, WIMG);
  const int rows = ey - sy, cols = ex - sx;

  if (threadIdx.x < 32) {
    const unsigned char* src = lh + ((size_t)n * HIMG + sy) * (size_t)WIMG + sx;
    tdm_load_2d_to_lds(inB, src, (unsigned)cols, (unsigned)rows, WIMG, 0u);
    __builtin_amdgcn_s_wait_tensorcnt(0);
  }
  __syncthreads();

  for (int i = threadIdx.x; i < HB * WB; i += 256) {
    const int ly = i / WB, lx = i % WB;
    const int gy = y0 - 2 + ly, gx = x0 - 2 + lx;
    unsigned v = 0;
    if (gy >= 0 && gy < HIMG && gx >= 0 && gx < WIMG)
      v = inB[(gy - sy) * cols + (gx - sx)];
    E0[i] = (unsigned char)(v & 1u);
    L0[i] = (unsigned char)((v >> 1) & 1u);
  }
  __syncthreads();

  // iteration 1 over tile + 1 halo ring
  for (int i = threadIdx.x; i < (HB - 2) * (WB - 2); i += 256) {
    const int ly = 1 + i / (WB - 2), lx = 1 + i % (WB - 2);
    const int p = ly * WB + lx;
    int any = 0;
#pragma unroll
    for (int dy = -1; dy <= 1; ++dy)
#pragma unroll
      for (int dx = -1; dx <= 1; ++dx) any |= E0[p + dy * WB + dx];
    const int conn = any & L0[p];
    E1[p] = (unsigned char)(E0[p] | conn);
    L1[p] = (unsigned char)(L0[p] & (conn ^ 1));
  }
  __syncthreads();

  // iteration 2 + clip + *255 -> output
  for (int i = threadIdx.x; i < TH * TW; i += 256) {
    const int ty = i / TW, tx = i % TW;
    const int gy = y0 + ty, gx = x0 + tx;
    if (gy >= HIMG || gx >= WIMG) continue;
    const int p = (ty + 2) * WB + (tx + 2);
    int any = 0;
#pragma unroll
    for (int dy = -1; dy <= 1; ++dy)
#pragma unroll
      for (int dx = -1; dx <= 1; ++dx) any |= E1[p + dy * WB + dx];
    const int conn = any & L1[p];
    const int e = (E1[p] | conn) & 1;
    edges[((size_t)n * HIMG + gy) * (size_t)WIMG + gx] = 255.0f * (float)e;
  }
}

// ---------------------------------------------------------------------------
extern "C" void kernel_launch(void* const* d_in, const int* in_sizes, int n_in,
                              void* d_out, int out_size, void* d_ws, size_t ws_size,
                              hipStream_t stream)
{
  const float* x   = (const float*)d_in[0];
  const float* gw  = (const float*)d_in[1];
  const float* sxw = (const float*)d_in[2];
  const float* syw = (const float*)d_in[3];

  const int P = HIMG * WIMG;
  const int N = in_sizes[0] / P;
  const int ntot = N * P;

  float* out   = (float*)d_out;
  float* edges = out;                       // output 0: edges (*255)
  float* magO  = out + (size_t)ntot;        // output 1: mag*255
  float* angO  = out + 2 * (size_t)ntot;    // output 2: angle deg % 180
  unsigned char* lh = (unsigned char*)d_ws; // N*P bytes of packed low/high

  dim3 tgrid((WIMG + TW - 1) / TW, (HIMG + TH - 1) / TH, N);
  canny_gauss_sobel<<<tgrid, 256, 0, stream>>>(x, gw, sxw, syw, magO, angO);

  const int blocksC = (ntot + 255) / 256;
  canny_nms<<<blocksC, 256, 0, stream>>>(magO, angO, lh, ntot);

  canny_hyst<<<tgrid, 256, 0, stream>>>(lh, edges);
}